// GNNGUARD_38628935860954
// MI455X (gfx1250) — compile-verified
//
#include <hip/hip_runtime.h>
#include <hip/hip_bf16.h>

typedef __attribute__((ext_vector_type(2))) float v2f;
typedef __attribute__((ext_vector_type(8))) float v8f;

#define D_FEAT 128
#define THRESH 0.1f
#define NORM_EPS 1e-8f

// ---------------------------------------------------------------------------
// Kernel 1: reciprocal clamped L2 norm per node. One wave32 per node:
// 32 lanes x float4 = 128 floats (one global_load_b128 per lane).
// Also serves to pull all of feat into the 192MB L2 before the gather kernel.
// ---------------------------------------------------------------------------
__global__ __launch_bounds__(256) void node_rnorm_kernel(const float* __restrict__ feat,
                                                         float* __restrict__ rnorm,
                                                         int n_nodes) {
    const int lane = threadIdx.x & 31;
    const int wave = blockIdx.x * (blockDim.x >> 5) + (threadIdx.x >> 5);
    if (wave >= n_nodes) return;
    const float4* p = (const float4*)(feat + (size_t)wave * D_FEAT);
    float4 v = p[lane];
    float ss = v.x * v.x + v.y * v.y + v.z * v.z + v.w * v.w;
    #pragma unroll
    for (int off = 16; off > 0; off >>= 1)
        ss += __shfl_xor(ss, off, 32);
    if (lane == 0)
        rnorm[wave] = 1.0f / fmaxf(sqrtf(ss), NORM_EPS);
}

// ---------------------------------------------------------------------------
// Kernel 2: batched edge dot products via V_WMMA_F32_16X16X4_F32.
// Each wave handles 16 edges. A[m][k] = feat[row[m]][k], B[k][n] = feat[col[n]][k].
// Accumulate C (16x16 f32) over 32 chunks of K=4; diag(C) = the 16 edge dots.
//
// ISA layouts (CDNA5 ISA):
//   A 16x4 f32:  lane m   -> VGPR0 = A[m][0], VGPR1 = A[m][1]
//                lane m+16-> VGPR0 = A[m][2], VGPR1 = A[m][3]
//     => lane loads contiguous float2 at k = 4*chunk + 2*(lane>>4)
//   B 4x16 f32:  row-striped across lanes, same K split (symmetric form)
//   C 16x16 f32: VGPR r: lanes 0-15 -> (M=r, N=lane); lanes 16-31 -> (M=r+8, N=lane-16)
//     => diag(i,i): i<8 at (vgpr=i, lane=i); i>=8 at (vgpr=i-8, lane=i+16)
// ---------------------------------------------------------------------------
__global__ __launch_bounds__(256) void edge_dot_wmma_kernel(const float* __restrict__ feat,
                                                            const int* __restrict__ row,
                                                            const int* __restrict__ col,
                                                            const float* __restrict__ rnorm,
                                                            float* __restrict__ att,
                                                            float* __restrict__ denom,
                                                            int n_edges) {
    const int lane  = threadIdx.x & 31;
    const int wave  = blockIdx.x * (blockDim.x >> 5) + (threadIdx.x >> 5);
    const int m     = lane & 15;      // edge-in-group this lane serves (A row / B col)
    const int khalf = lane >> 4;      // which K pair within the chunk

    const int eb = wave * 16;
    if (eb >= n_edges) return;

    const int e  = eb + m;
    const int ec = (e < n_edges) ? e : (n_edges - 1);   // clamp tail (dup work, masked store)
    const int rI = row[ec];
    const int cI = col[ec];
    const float* arow = feat + (size_t)rI * D_FEAT;
    const float* brow = feat + (size_t)cI * D_FEAT;

    v8f acc = {};
    #pragma unroll
    for (int ch = 0; ch < D_FEAT / 4; ++ch) {
        const int kb = ch * 4 + khalf * 2;
        v2f a, b;
        a.x = arow[kb];
        a.y = arow[kb + 1];
        b.x = brow[kb];
        b.y = brow[kb + 1];
        acc = __builtin_amdgcn_wmma_f32_16x16x4_f32(
            /*neg_a=*/false, a, /*neg_b=*/false, b,
            /*c_mod=*/(short)0, acc, /*reuse_a=*/false, /*reuse_b=*/false);
    }

    // Extract diag(C): edge i lives at (vgpr=i, lane=i) for i<8,
    // (vgpr=i-8, lane=(i-8)+24 = i+16) for i>=8. Distribute to lane i.
    float dot = 0.0f;
    #pragma unroll
    for (int r = 0; r < 8; ++r) {
        float t  = acc[r];
        float lo = __shfl(t, r, 32);        // diag element r
        float hi = __shfl(t, r + 24, 32);   // diag element r+8
        if (lane == r)      dot = lo;
        if (lane == r + 8)  dot = hi;
    }

    if (lane < 16 && e < n_edges) {
        float a = dot * rnorm[rI] * rnorm[cI];   // cosine similarity
        a = (a < THRESH) ? 0.0f : a;             // threshold (result nonnegative)
        att[e] = a;
        if (a > 0.0f)
            atomicAdd(denom + rI, a);            // per-source-row L1 denom
    }
}

// ---------------------------------------------------------------------------
// Kernel 3: out[e] = exp( denom[row[e]] > 0 ? att[e]/denom[row[e]] : 0 )
// ---------------------------------------------------------------------------
__global__ __launch_bounds__(256) void finalize_kernel(const float* __restrict__ att,
                                                       const float* __restrict__ denom,
                                                       const int* __restrict__ row,
                                                       float* __restrict__ out,
                                                       int n_edges) {
    const int e = blockIdx.x * blockDim.x + threadIdx.x;
    if (e >= n_edges) return;
    const float a  = att[e];
    const float dn = denom[row[e]];
    const float an = (dn > 0.0f) ? (a / dn) : 0.0f;
    out[e] = expf(an);
}

extern "C" void kernel_launch(void* const* d_in, const int* in_sizes, int n_in,
                              void* d_out, int out_size, void* d_ws, size_t ws_size,
                              hipStream_t stream) {
    const float* feat = (const float*)d_in[0];
    const int*   row  = (const int*)d_in[1];
    const int*   col  = (const int*)d_in[2];
    float*       out  = (float*)d_out;

    const int n_nodes = in_sizes[0] / D_FEAT;
    const int n_edges = in_sizes[1];

    // workspace layout: [rnorm: n_nodes][denom: n_nodes][att: n_edges] floats
    float* rnorm = (float*)d_ws;
    float* denom = rnorm + n_nodes;
    float* att   = denom + n_nodes;

    hipMemsetAsync(denom, 0, (size_t)n_nodes * sizeof(float), stream);

    {   // kernel 1: 8 waves (8 nodes) per 256-thread block
        int blocks = (n_nodes + 7) / 8;
        node_rnorm_kernel<<<blocks, 256, 0, stream>>>(feat, rnorm, n_nodes);
    }
    {   // kernel 2: 16 edges per wave, 8 waves per block
        int n_waves = (n_edges + 15) / 16;
        int blocks  = (n_waves + 7) / 8;
        edge_dot_wmma_kernel<<<blocks, 256, 0, stream>>>(feat, row, col, rnorm,
                                                         att, denom, n_edges);
    }
    {   // kernel 3: elementwise
        int blocks = (n_edges + 255) / 256;
        finalize_kernel<<<blocks, 256, 0, stream>>>(att, denom, row, out, n_edges);
    }
}